// MRGE_bert_56160992362626
// MI455X (gfx1250) — compile-verified
//
#include <hip/hip_runtime.h>

// Problem constants (match reference)
#define B_    32
#define S_    1024
#define NODE_ 256
#define DEP_  64
#define R_    50
#define L_    16
#define E_    128
#define F_    (NODE_ + DEP_)   // 320
#define NE_   (B_ * E_)        // 4096 edges per layer

typedef __attribute__((ext_vector_type(2))) float v2f;
typedef __attribute__((ext_vector_type(8))) float v8f;

// ---------------------------------------------------------------------------
// Per-layer relation-grouped GEMM:  msg[64 x Ne] = W_r[64x320] * Feat[320 x Ne]
// One block per relation r; 128 threads = 4 waves; wave w computes M-tile w.
// Uses V_WMMA_F32_16X16X4_F32 (exact fp32, matches reference precision).
// ---------------------------------------------------------------------------
__global__ __launch_bounds__(128) void layer_gemm_kernel(
    const float* __restrict__ context,   // [B,S,NODE]
    const float* __restrict__ depW,      // [R,DEP,F]
    const int*   __restrict__ heads,     // [B,L,E]
    const int*   __restrict__ tails,     // [B,L,E]
    const int*   __restrict__ rels,      // [B,L,E]
    const float* __restrict__ child,     // [B*S,DEP]   (state from previous layer)
    float*       __restrict__ seg,       // [B*S,DEP]   (message accumulator, pre-zeroed)
    float*       __restrict__ cnt,       // [B*S]       (edge counts, pre-zeroed)
    int layer)
{
    __shared__ int   s_list[NE_];        // edge ids (flat b*E+e) with rel==r
    __shared__ int   s_num;
    __shared__ float s_feat[F_ * 16];    // Feat tile, [k][n] layout
    __shared__ int   s_head[16];
    __shared__ int   s_batch[16];

    const int r   = blockIdx.x;
    const int tid = threadIdx.x;

    if (tid == 0) s_num = 0;
    __syncthreads();

    // ---- build edge list for this relation (and bump head counts) ----
    for (int i = tid; i < NE_; i += 128) {
        const int b   = i >> 7;
        const int e   = i & (E_ - 1);
        const int idx = (b * L_ + layer) * E_ + e;
        if (rels[idx] == r) {
            const int p = atomicAdd(&s_num, 1);
            s_list[p] = i;
            atomicAdd(&cnt[b * S_ + heads[idx]], 1.0f);
        }
    }
    __syncthreads();

    const int num   = s_num;
    const int lane  = tid & 31;
    const int wave  = tid >> 5;          // M-tile index 0..3
    const int nloc  = lane & 15;         // N column within tile / A row within M-tile
    const int kbase = (lane < 16) ? 0 : 2;

    const float* Wr   = depW + (size_t)r * DEP_ * F_;
    const float* Arow = Wr + (size_t)(wave * 16 + nloc) * F_;

    for (int t0 = 0; t0 < num; t0 += 16) {
        const int tcnt = min(16, num - t0);

        // ---- gather feat tile: 8 threads per edge column ----
        {
            const int n  = tid >> 3;                       // 0..15 edge column
            const int p  = tid & 7;                        // 0..7 chunk
            const int li = t0 + ((n < tcnt) ? n : (tcnt - 1));  // clamp-pad
            const int i  = s_list[li];
            const int b  = i >> 7;
            const int e  = i & (E_ - 1);
            const int idx   = (b * L_ + layer) * E_ + e;
            const int tnode = tails[idx];
            if (p == 0) { s_head[n] = heads[idx]; s_batch[n] = b; }
            const float* cx = context + ((size_t)b * S_ + tnode) * NODE_;
            const float* ch = child   + ((size_t)b * S_ + tnode) * DEP_;
            #pragma unroll 4
            for (int k = p * 32; k < p * 32 + 32; ++k)
                s_feat[k * 16 + n] = cx[k];
            #pragma unroll
            for (int j = 0; j < 8; ++j)
                s_feat[(NODE_ + p * 8 + j) * 16 + n] = ch[p * 8 + j];
        }
        __syncthreads();

        // ---- fp32 WMMA K-loop: acc(16x16) += A(16x4) * B(4x16) ----
        v8f acc = {};
        for (int k0 = 0; k0 < F_; k0 += 4) {
            v2f a, bb;
            a.x  = Arow[k0 + kbase];
            a.y  = Arow[k0 + kbase + 1];
            bb.x = s_feat[(k0 + kbase)     * 16 + nloc];
            bb.y = s_feat[(k0 + kbase + 1) * 16 + nloc];
            acc = __builtin_amdgcn_wmma_f32_16x16x4_f32(
                /*neg_a=*/false, a, /*neg_b=*/false, bb,
                /*c_mod=*/(short)0, acc, /*reuse_a=*/false, /*reuse_b=*/false);
        }

        // ---- scatter-add message tile onto head nodes ----
        if (nloc < tcnt) {
            const int h = s_head[nloc];
            const int b = s_batch[nloc];
            const size_t base = ((size_t)b * S_ + h) * DEP_
                              + wave * 16 + ((lane < 16) ? 0 : 8);
            #pragma unroll
            for (int j = 0; j < 8; ++j)
                atomicAdd(&seg[base + j], acc[j]);
        }
        __syncthreads();   // protect s_feat before next gather
    }
}

// ---------------------------------------------------------------------------
// child = cnt>0 ? seg/cnt : child ; reset seg/cnt for next layer.
// Grid covers B*S*DEP exactly; each node's 64 lanes live inside one block.
// ---------------------------------------------------------------------------
__global__ __launch_bounds__(256) void finalize_kernel(
    float* __restrict__ child, float* __restrict__ seg, float* __restrict__ cnt)
{
    const int i  = blockIdx.x * 256 + threadIdx.x;   // [0, B*S*DEP)
    const int bs = i / DEP_;
    const float c = cnt[bs];
    if (c > 0.0f) child[i] = seg[i] / c;
    seg[i] = 0.0f;
    __syncthreads();                                  // all reads of cnt done
    if ((i & (DEP_ - 1)) == 0) cnt[bs] = 0.0f;
}

// ---------------------------------------------------------------------------
// out[b,s,:] = context[b,s,:] ++ child[b,s,:]
// ---------------------------------------------------------------------------
__global__ __launch_bounds__(256) void output_kernel(
    const float* __restrict__ context, const float* __restrict__ child,
    float* __restrict__ out)
{
    const size_t i  = (size_t)blockIdx.x * 256 + threadIdx.x;  // [0, B*S*F)
    const size_t bs = i / F_;
    const int    ch = (int)(i % F_);
    out[i] = (ch < NODE_) ? context[bs * NODE_ + ch]
                          : child[bs * DEP_ + (ch - NODE_)];
}

extern "C" void kernel_launch(void* const* d_in, const int* in_sizes, int n_in,
                              void* d_out, int out_size, void* d_ws, size_t ws_size,
                              hipStream_t stream) {
    const float* context = (const float*)d_in[0];
    const float* depW    = (const float*)d_in[1];
    const int*   heads   = (const int*)d_in[2];
    const int*   tails   = (const int*)d_in[3];
    const int*   rels    = (const int*)d_in[4];
    float*       out     = (float*)d_out;

    float* child = (float*)d_ws;                        // [B*S, DEP]  8 MB
    float* seg   = child + (size_t)B_ * S_ * DEP_;      // [B*S, DEP]  8 MB
    float* cnt   = seg   + (size_t)B_ * S_ * DEP_;      // [B*S]       128 KB

    const size_t zbytes =
        ((size_t)B_ * S_ * DEP_ * 2 + (size_t)B_ * S_) * sizeof(float);
    hipMemsetAsync(d_ws, 0, zbytes, stream);            // capture-legal

    for (int l = 0; l < L_; ++l) {
        layer_gemm_kernel<<<R_, 128, 0, stream>>>(
            context, depW, heads, tails, rels, child, seg, cnt, l);
        finalize_kernel<<<(B_ * S_ * DEP_) / 256, 256, 0, stream>>>(child, seg, cnt);
    }
    output_kernel<<<(int)(((size_t)B_ * S_ * F_) / 256), 256, 0, stream>>>(
        context, child, out);
}